// anchor2anchor_SimLoss_75685913690504
// MI455X (gfx1250) — compile-verified
//
#include <hip/hip_runtime.h>
#include <hip/hip_bf16.h>
#include <math.h>
#include <stdint.h>

typedef __attribute__((ext_vector_type(2))) float v2f;
typedef __attribute__((ext_vector_type(8))) float v8f;
typedef int v4i __attribute__((vector_size(16)));   // 128-bit payload type

#define B_ROWS 8192
#define DIM    128
#define NTILE  512              // 8192 / 16 j-tiles
#define SCALE  10.0f            // 2 / T, T = 0.2

// sim_loss tiling
#define ITILES   2              // i-tiles per block (32 rows)
#define JSLOTS   4              // j-tiles staged per stage
#define NSTAGE   (NTILE / JSLOTS)      // 128 stages
#define SROWS    (JSLOTS * 16)         // 64 Z-rows staged per stage
#define LDSTRIDE 132            // dwords per staged row (128 + 4 pad: 16B-aligned,
                                // bank = (4n + c) % 64 -> conflict-free ds_load_b64)

// ---------------------------------------------------------------------------
// CDNA5 async global->LDS copy (ASYNCcnt) with inline-asm fallback.
// ---------------------------------------------------------------------------
__device__ __forceinline__ void async_b128(const float* g, float* l) {
#if __has_builtin(__builtin_amdgcn_global_load_async_to_lds_b128)
    __attribute__((address_space(1))) v4i* gp =
        (__attribute__((address_space(1))) v4i*)(uintptr_t)g;
    __attribute__((address_space(3))) v4i* lp =
        (__attribute__((address_space(3))) v4i*)(unsigned)(uintptr_t)l;
    __builtin_amdgcn_global_load_async_to_lds_b128(gp, lp, 0, 0);
#else
    const unsigned lds_off = (unsigned)(uintptr_t)l;   // generic LDS ptr low 32b = LDS offset
    asm volatile("global_load_async_to_lds_b128 %0, %1, off"
                 :: "v"(lds_off), "v"(g)
                 : "memory");
#endif
}

#if __has_builtin(__builtin_amdgcn_s_wait_asynccnt)
#define WAIT_ASYNC(n) __builtin_amdgcn_s_wait_asynccnt(n)
#else
#define WAIT_ASYNC(n) asm volatile("s_wait_asynccnt %0" :: "i"(n) : "memory")
#endif

// ---------------------------------------------------------------------------
// Kernel 0: zero the scalar output (harness poisons d_out with 0xAA).
// ---------------------------------------------------------------------------
__global__ void zero_out_kernel(float* out) { out[0] = 0.0f; }

// ---------------------------------------------------------------------------
// Kernel 1: z = A @ W + bias, then L2-normalize each row.  One 128-thread
// block per row.  ~0.27 GFLOP total: VALU is fine.
// ---------------------------------------------------------------------------
__global__ __launch_bounds__(128)
void proj_norm_kernel(const float* __restrict__ A,
                      const float* __restrict__ W,
                      const float* __restrict__ bias,
                      float* __restrict__ Z) {
    __shared__ float arow[DIM];
    __shared__ float red[DIM];
    const int i = blockIdx.x;
    const int t = threadIdx.x;

    arow[t] = A[i * DIM + t];
    __syncthreads();

    float acc = bias[t];
#pragma unroll 4
    for (int k = 0; k < DIM; ++k)
        acc = fmaf(arow[k], W[k * DIM + t], acc);

    red[t] = acc * acc;
    __syncthreads();
#pragma unroll
    for (int s = 64; s > 0; s >>= 1) {
        if (t < s) red[t] += red[t + s];
        __syncthreads();
    }
    const float norm = fmaxf(sqrtf(red[0]), 1e-12f);
    Z[i * DIM + t] = acc / norm;
}

// ---------------------------------------------------------------------------
// Kernel 2: fused Gram matrix + contrastive log-sum-exp reduction.
//
// 256 blocks; block owns 2 i-tiles (32 rows).  8 waves: wave w ->
// (isub = w & 1, jslot = w >> 1 in 0..3).  Stage s async-copies Z rows
// [64s, 64s+64) (4 j-tiles, 32 KB contiguous) into LDS (double-buffered,
// ASYNCcnt).  Each wave then runs 32x V_WMMA_F32_16X16X4_F32 for its
// (i-tile, j-tile = 4s + jslot) pair: A-fragments preloaded in 64 VGPRs,
// B-fragments from LDS via conflict-free ds_load_b64.
//
// Fragment gather (ISA 7.12.2, 32-bit 16x4 / 4x16): lane l -> row l&15,
// K = 4*kk + 2*(l>>4) .. +1.  C/D: VGPR v, lane l -> (M=v+8*(l>>4), N=l&15).
// ---------------------------------------------------------------------------
__global__ __launch_bounds__(256)
void sim_loss_kernel(const float* __restrict__ Z, float* __restrict__ out) {
    __shared__ float buf[2][SROWS * LDSTRIDE];      // 2 x 33.8 KB staging
    __shared__ float sB[ITILES][16][JSLOTS];
    __shared__ float sT[ITILES][16][JSLOTS];
    __shared__ float sTerm[ITILES * 16];

    const int tid   = threadIdx.x;
    const int wave  = tid >> 5;
    const int lane  = tid & 31;
    const int isub  = wave & 1;
    const int jslot = wave >> 1;                    // 0..3
    const int m     = lane & 15;
    const int hi    = lane >> 4;
    const int ksel  = hi << 1;                      // 0 or 2

    const int it    = blockIdx.x * ITILES + isub;   // global i-tile (0..511)
    const int jt0   = (it + NTILE / 2) & (NTILE - 1);

    // Preload A-matrix fragments for this wave's i-tile (K = 0..127).
    const float* zi = Z + (it * 16 + m) * DIM + ksel;
    v2f av[32];
#pragma unroll
    for (int kk = 0; kk < 32; ++kk)
        av[kk] = *(const v2f*)(zi + kk * 4);

    float bottomAcc[8];
    float topAcc[8];
#pragma unroll
    for (int v = 0; v < 8; ++v) { bottomAcc[v] = 0.0f; topAcc[v] = 0.0f; }

    // --- async staging: thread copies rows (wave + 8q), chunk = lane -------
    // stage s -> global rows g0 = 64*s, 32 b128 chunks per row, 8 per thread.
    {
        float* dst = &buf[0][0];
        const float* src = Z;                       // stage 0, g0 = 0
#pragma unroll
        for (int q = 0; q < 8; ++q) {
            const int r = wave + 8 * q;
            async_b128(src + r * DIM + lane * 4, dst + r * LDSTRIDE + lane * 4);
        }
    }

    for (int s = 0; s < NSTAGE; ++s) {
        const int p = s & 1;
        if (s + 1 < NSTAGE) {
            float* dst = &buf[(s + 1) & 1][0];
            const float* src = Z + (s + 1) * SROWS * DIM;
#pragma unroll
            for (int q = 0; q < 8; ++q) {
                const int r = wave + 8 * q;
                async_b128(src + r * DIM + lane * 4, dst + r * LDSTRIDE + lane * 4);
            }
            WAIT_ASYNC(8);        // stage s's 8 copies (issued earlier) done
        } else {
            WAIT_ASYNC(0);
        }
        __syncthreads();          // all waves' copies for stage s visible

        // ---- compute tile (it, jt) from LDS ------------------------------
        const int jt = s * JSLOTS + jslot;
        const float* bbase = &buf[p][(jslot * 16 + m) * LDSTRIDE + ksel];

        v8f acc = {};
#pragma unroll
        for (int kk = 0; kk < 32; ++kk) {
            v2f bv = *(const v2f*)(bbase + kk * 4);
            acc = __builtin_amdgcn_wmma_f32_16x16x4_f32(
                false, av[kk], false, bv, (short)0, acc, false, false);
        }

        const bool diagTile = (jt == it);
        const bool topTile  = (jt == jt0);
#pragma unroll
        for (int v = 0; v < 8; ++v) {
            const float sv = SCALE * acc[v];
            const bool isDiagSlot = ((lane & 15) == (v + (hi << 3)));
            const float e = __expf(sv);
            if (!(diagTile && isDiagSlot)) bottomAcc[v] += e;   // sum_{j != i}
            if (topTile && isDiagSlot)     topAcc[v]    = sv;   // SCALE * d[i,j0]
        }
        __syncthreads();          // done reading buf[p] before it is refilled
    }

    // Reduce over the 16 lanes holding one row (wave32 halves independent).
#pragma unroll
    for (int v = 0; v < 8; ++v) {
#pragma unroll
        for (int off = 1; off < 16; off <<= 1) {
            bottomAcc[v] += __shfl_xor(bottomAcc[v], off, 32);
            topAcc[v]    += __shfl_xor(topAcc[v],    off, 32);
        }
    }
    // lane 0 holds rows M=0..7 (index v), lane 16 holds M=8..15.
    if (lane == 0) {
#pragma unroll
        for (int v = 0; v < 8; ++v) {
            sB[isub][v][jslot] = bottomAcc[v];
            sT[isub][v][jslot] = topAcc[v];
        }
    } else if (lane == 16) {
#pragma unroll
        for (int v = 0; v < 8; ++v) {
            sB[isub][v + 8][jslot] = bottomAcc[v];
            sT[isub][v + 8][jslot] = topAcc[v];
        }
    }
    __syncthreads();

    if (tid < ITILES * 16) {
        const int is = tid >> 4, r = tid & 15;
        float bsum = 0.0f, tsum = 0.0f;
#pragma unroll
        for (int q = 0; q < JSLOTS; ++q) {
            bsum += sB[is][r][q];
            tsum += sT[is][r][q];
        }
        sTerm[tid] = tsum - __logf(bsum);           // log(top) - log(bottom)
    }
    __syncthreads();

    if (tid == 0) {
        float blockSum = 0.0f;
#pragma unroll
        for (int r = 0; r < ITILES * 16; ++r) blockSum += sTerm[r];
        atomicAdd(out, -blockSum / (float)(B_ROWS - 1));
    }
}

// ---------------------------------------------------------------------------
// Launch
// ---------------------------------------------------------------------------
extern "C" void kernel_launch(void* const* d_in, const int* in_sizes, int n_in,
                              void* d_out, int out_size, void* d_ws, size_t ws_size,
                              hipStream_t stream) {
    const float* A    = (const float*)d_in[0];   // [8192, 128]
    const float* W    = (const float*)d_in[1];   // [128, 128]
    const float* bias = (const float*)d_in[2];   // [128]
    float* out        = (float*)d_out;           // scalar loss
    float* Z          = (float*)d_ws;            // [8192, 128] normalized

    zero_out_kernel<<<1, 1, 0, stream>>>(out);
    proj_norm_kernel<<<B_ROWS, DIM, 0, stream>>>(A, W, bias, Z);
    sim_loss_kernel<<<NTILE / ITILES, 256, 0, stream>>>(Z, out);
}